// PianoSSM_XL_35467839930784
// MI455X (gfx1250) — compile-verified
//
#include <hip/hip_runtime.h>
#include <cstddef>

// ---------------- problem constants ----------------
#define PDIM 256              // SSM state size P (all layers)
#define PP   260              // padded LDS row stride: ==4 (mod 64) -> aligned
                              // conflict-free ds_load_b64 A-fragments
#define TCH  64               // timesteps per chunk (per workgroup)
#define LSEQ 16384            // 512 * COPY(32)
#define NCHUNK (LSEQ / TCH)   // 256 chunks per batch
#define NB   8                // batch

typedef __attribute__((ext_vector_type(2))) float v2f;
typedef __attribute__((ext_vector_type(8))) float v8f;

__device__ __forceinline__ v8f wmma4(v2f a, v2f b, v8f c) {
  // V_WMMA_F32_16X16X4_F32 : D(16x16,f32) = A(16x4,f32) x B(4x16,f32) + C
  return __builtin_amdgcn_wmma_f32_16x16x4_f32(false, a, false, b, (short)0, c,
                                               false, false);
}

__device__ __forceinline__ v8f zero8() {
  v8f z = {0.f, 0.f, 0.f, 0.f, 0.f, 0.f, 0.f, 0.f};
  return z;
}

__device__ __forceinline__ v2f ld2s(const float* p) {  // aligned LDS/global pair
  return *(const v2f*)p;
}

// ZOH discretization: lam_bar = exp((lre + i*lim) * exp(log_step))
__device__ __forceinline__ void disc(float lre, float lim, float ls,
                                     float& lbr, float& lbi) {
  float dt = expf(ls);
  float m = expf(lre * dt);
  float s, c;
  sincosf(lim * dt, &s, &c);
  lbr = m * c;
  lbi = m * s;
}

// ---------------- Tensor Data Mover staging (CDNA5) ----------------
#if __has_builtin(__builtin_amdgcn_tensor_load_to_lds) && \
    __has_builtin(__builtin_amdgcn_s_wait_tensorcnt)
#define USE_TDM 1
#else
#define USE_TDM 0
#endif

typedef __attribute__((ext_vector_type(4))) unsigned tdm_v4u;
typedef __attribute__((ext_vector_type(8))) int tdm_v8i;
typedef __attribute__((ext_vector_type(4))) int tdm_v4i;

__device__ __forceinline__ unsigned lds_addr_of(const void* p) {
  // AS(3) pointer value == byte offset within the workgroup's LDS allocation
  return (unsigned)(unsigned long long)(
      __attribute__((address_space(3))) const void*)p;
}

#if USE_TDM
// Copy nrows rows of rowFloats f32 from global (row step gIncFloats) to LDS
// (row step ldsStrideFloats) using one TDM descriptor in iterate mode.
__device__ __forceinline__ void tdm_load_rows(const float* gsrc,
                                              unsigned ldsByte, int rowFloats,
                                              int ldsStrideFloats, int nrows,
                                              int gIncFloats) {
  unsigned long long ga = (unsigned long long)gsrc;
  tdm_v4u g0;
  g0.x = 1u;                                    // count=1, user descriptor
  g0.y = ldsByte;                               // lds_addr
  g0.z = (unsigned)(ga & 0xffffffffull);        // global_addr[31:0]
  g0.w = (unsigned)((ga >> 32) & 0x01ffffffull) // global_addr[56:32]
         | 0x80000000u;                         // type=2 ("image")
  tdm_v8i g1;
  g1[0] = (2 << 16) | (1 << 19);                // data_size=4B, iterate_enable
  g1[1] = (rowFloats & 0xffff) << 16;           // tensor_dim0[15:0]
  g1[2] = (nrows & 0xffff) << 16;               // dim0 hi=0 | tensor_dim1 lo
  g1[3] = (rowFloats & 0xffff) << 16;           // dim1 hi=0 | tile_dim0
  g1[4] = 1;                                    // tile_dim1=1, tile_dim2=0
  g1[5] = rowFloats;                            // tensor_dim0_stride lo32
  g1[6] = 0;
  g1[7] = 0;
  tdm_v4i g2;
  g2[0] = 1;                                    // tensor_dim2 (unused)
  g2[1] = ldsStrideFloats;                      // lds_addr_increment / iter
  g2[2] = gIncFloats;                           // global_addr_increment lo
  g2[3] = (nrows - 1) << 16;                    // iterate_count (0 => 1x)
  tdm_v4i g3 = {0, 0, 0, 0};
  tdm_v8i g4 = {0, 0, 0, 0, 0, 0, 0, 0};        // upper groups (unused, 2D)
  __builtin_amdgcn_tensor_load_to_lds(g0, g1, g2, g3, g4, 0);
}
#endif

// ---------------- per-layer parameter prep ----------------
// Packed "fragment" weight layouts: pair index q = k*2+half selects the two
// consecutive K-values a lane consumes -> one global_load_b64 per fragment.
//   BtPRe/Im [H/2][256][2]  : B_bar^T    (Bu GEMM operand B)
//   CtPRe    [128][OPAD][2] : 2*C_re^T   ; CtPIm = -2*C_im^T (padded cols)
//   skipWP   [H/2][OPAD][2] ; boutP/skipBp [OPAD]
//   lamRe/Im [256]=lam_bar ; lamTRe/Im [256]=lam_bar^TCH
//   binBtRe/Im [256] = b_in @ B_bar^T  (constant per-channel Bu offset)
__global__ void prep_kernel(
    const float* __restrict__ B_im, const float* __restrict__ B_re,
    const float* __restrict__ C_im, const float* __restrict__ C_re,
    const float* __restrict__ b_out,
    const float* __restrict__ lam_im, const float* __restrict__ lam_re,
    const float* __restrict__ log_step,
    const float* __restrict__ skip_W, const float* __restrict__ skip_b,
    const float* __restrict__ b_in,
    float* __restrict__ BtPRe, float* __restrict__ BtPIm,
    float* __restrict__ CtPRe, float* __restrict__ CtPIm,
    float* __restrict__ skipWP, float* __restrict__ boutP,
    float* __restrict__ skipBp,
    float* __restrict__ binBtRe, float* __restrict__ binBtIm,
    float* __restrict__ lamRe, float* __restrict__ lamIm,
    float* __restrict__ lamTRe, float* __restrict__ lamTIm,
    int H, int O, int OPAD, int hasSkip) {
  int idx = blockIdx.x * blockDim.x + threadIdx.x;
  const int nLam = PDIM;
  const int nBt = H * PDIM;
  const int nCt = PDIM * OPAD;
  const int nSk = H * OPAD;
  const int nBo = OPAD;

  if (idx < nLam) {
    int p = idx;
    float lbr, lbi;
    disc(lam_re[p], lam_im[p], log_step[p], lbr, lbi);
    lamRe[p] = lbr;
    lamIm[p] = lbi;
    float zr = lbr, zi = lbi;  // lam_bar^(2^k); TCH = 64 = 2^6
    for (int i = 0; i < 6; ++i) {
      float nr = zr * zr - zi * zi;
      float ni = 2.f * zr * zi;
      zr = nr; zi = ni;
    }
    lamTRe[p] = zr;
    lamTIm[p] = zi;
    return;
  }
  idx -= nLam;
  if (idx < nBt) {
    int h = idx / PDIM, p = idx - (idx / PDIM) * PDIM;
    float lre = lam_re[p], lim = lam_im[p];
    float lbr, lbi;
    disc(lre, lim, log_step[p], lbr, lbi);
    float den = lre * lre + lim * lim;
    float wr = ((lbr - 1.f) * lre + lbi * lim) / den;  // (lam_bar-1)/lam
    float wi = (lbi * lre - (lbr - 1.f) * lim) / den;
    float br = B_re[p * H + h], bi = B_im[p * H + h];
    int dst = (h >> 1) * (PDIM * 2) + p * 2 + (h & 1);
    BtPRe[dst] = wr * br - wi * bi;
    BtPIm[dst] = wr * bi + wi * br;
    return;
  }
  idx -= nBt;
  if (idx < nCt) {
    int p = idx / OPAD, o = idx - (idx / OPAD) * OPAD;
    float cr = 0.f, ci = 0.f;
    if (o < O) {
      cr = 2.f * C_re[o * PDIM + p];
      ci = -2.f * C_im[o * PDIM + p];
    }
    int dst = (p >> 1) * (OPAD * 2) + o * 2 + (p & 1);
    CtPRe[dst] = cr;
    CtPIm[dst] = ci;
    return;
  }
  idx -= nCt;
  if (idx < nSk) {
    int h = idx / OPAD, o = idx - (idx / OPAD) * OPAD;
    int dst = (h >> 1) * (OPAD * 2) + o * 2 + (h & 1);
    skipWP[dst] = (hasSkip && o < O) ? skip_W[h * O + o] : 0.f;
    return;
  }
  idx -= nSk;
  if (idx < nBo) {
    int o = idx;
    boutP[o] = (o < O) ? b_out[o] : 0.f;
    skipBp[o] = (hasSkip && o < O) ? skip_b[o] : 0.f;
    return;
  }
  idx -= nBo;
  if (idx < PDIM) {  // binBt[p] = sum_h b_in[h] * B_bar[p][h]
    int p = idx;
    float lre = lam_re[p], lim = lam_im[p];
    float lbr, lbi;
    disc(lre, lim, log_step[p], lbr, lbi);
    float den = lre * lre + lim * lim;
    float wr = ((lbr - 1.f) * lre + lbi * lim) / den;
    float wi = (lbi * lre - (lbr - 1.f) * lim) / den;
    float sr = 0.f, si = 0.f;
    for (int h = 0; h < H; ++h) {
      float bv = b_in[h];
      float br = B_re[p * H + h], bi = B_im[p * H + h];
      sr += bv * (wr * br - wi * bi);
      si += bv * (wr * bi + wi * br);
    }
    binBtRe[p] = sr;
    binBtIm[p] = si;
  }
}

// ---------------- fused SSM layer kernel ----------------
// One workgroup = one (batch b, chunk c of TCH timesteps):
//   1) TDM-stage raw u tile into LDS (float4 fallback)
//   2) Bu = u @ B_bar^T  (f32 WMMA, both complex planes) -> LDS [t][p]
//   3) local scan x_t = lam_bar*x_{t-1} + (Bu_t + binBt) (thread p = channel)
//   4) decoupled carry chain across chunks, then x_t += lam_bar^{t+1}*carry
//   5) y = xs_re@(2Cre^T) + xs_im@(-2Cim^T) + b_out; relu/sqrt2; + u@skipW
template <int H, int OT, bool SKIP, bool REPEAT_IN>
__global__ __launch_bounds__(256) void ssm_layer(
    const float* __restrict__ xin, float* __restrict__ xout,
    const float* __restrict__ BtPRe, const float* __restrict__ BtPIm,
    const float* __restrict__ CtPRe, const float* __restrict__ CtPIm,
    const float* __restrict__ skipWP, const float* __restrict__ boutP,
    const float* __restrict__ skipBp,
    const float* __restrict__ binBtRe, const float* __restrict__ binBtIm,
    const float* __restrict__ lamRe, const float* __restrict__ lamIm,
    const float* __restrict__ lamTRe, const float* __restrict__ lamTIm,
    float* __restrict__ carryRe, float* __restrict__ carryIm,
    int* __restrict__ flags, int O) {
  constexpr int HPAD = (H == 60) ? 68 : H + 4;  // 16B rows, bank-clean b64
  constexpr int KH = H / 4;                     // K-steps for WMMA 16x16x4
  constexpr int OPAD = OT * 16;
  static_assert(KH * 4 == H, "H must be a multiple of 4");

  __shared__ float s_re[TCH * PP];   // Bu / states, real plane
  __shared__ float s_im[TCH * PP];   // Bu / states, imag plane
  __shared__ float s_u[TCH * HPAD];  // raw input tile

  const int tid = threadIdx.x;  // 256 threads = 8 waves (wave32)
  const int wave = tid >> 5;
  const int lane = tid & 31;
  const int ln = lane & 15;
  const int hi = lane >> 4;     // half-wave (ISA fragment layouts)

  const int c = blockIdx.x >> 3;  // chunk (chain order == blockIdx order)
  const int b = blockIdx.x & 7;   // batch
  const int t0 = c * TCH;

  // ---- 1) stage raw input tile ----
#if USE_TDM
  if (wave == 0) {
    if (REPEAT_IN) {
      // repeat(midi, 32): two midi rows, each broadcast into 32 LDS rows
      const float* r0 = xin + ((size_t)b * 512 + (t0 >> 5)) * 88;
      tdm_load_rows(r0, lds_addr_of(s_u), H, HPAD, 32, 0);
      tdm_load_rows(r0 + 88, lds_addr_of(s_u) + 32u * HPAD * 4u, H, HPAD, 32,
                    0);
    } else {
      tdm_load_rows(xin + ((size_t)b * LSEQ + t0) * H, lds_addr_of(s_u), H,
                    HPAD, TCH, H);
    }
    __builtin_amdgcn_s_wait_tensorcnt(0);
  }
#else
  for (int i = tid; i < TCH * (H / 4); i += 256) {
    int t = i / (H / 4), seg = i - t * (H / 4);
    int tg = t0 + t;
    const float* src =
        REPEAT_IN ? xin + ((size_t)b * 512 + (tg >> 5)) * 88 + seg * 4
                  : xin + ((size_t)b * LSEQ + tg) * H + seg * 4;
    *(float4*)&s_u[t * HPAD + seg * 4] = *(const float4*)src;
  }
#endif
  __syncthreads();

  // ---- 2) Bu GEMM: columns 0..255 -> re plane, 256..511 -> im plane ----
  for (int nt = wave; nt < 32; nt += 8) {
    const float* __restrict__ BtP = (nt < 16) ? BtPRe : BtPIm;
    float* plane = (nt < 16) ? s_re : s_im;
    const int p0 = (nt & 15) * 16;
    v2f bf[KH];
#pragma unroll
    for (int k = 0; k < KH; ++k)  // one b64 per fragment (packed layout)
      bf[k] = ld2s(&BtP[(k * 2 + hi) * (PDIM * 2) + (p0 + ln) * 2]);
#pragma unroll
    for (int mt = 0; mt < TCH / 16; ++mt) {
      v8f acc = zero8();
#pragma unroll
      for (int k = 0; k < KH; ++k) {
        v2f a = ld2s(&s_u[(mt * 16 + ln) * HPAD + k * 4 + hi * 2]);
        acc = wmma4(a, bf[k], acc);
      }
      const int rb = mt * 16 + hi * 8;
#pragma unroll
      for (int r = 0; r < 8; ++r)
        plane[(rb + r) * PP + p0 + ln] = acc[r];
    }
  }
  __syncthreads();

  // ---- 3) local scan (zero init), thread tid owns channel p = tid ----
  const float lr = lamRe[tid], li = lamIm[tid];
  const float binR = binBtRe[tid], binI = binBtIm[tid];
  float xr = 0.f, xi = 0.f;
#pragma unroll 4
  for (int t = 0; t < TCH; ++t) {
    float br = s_re[t * PP + tid] + binR;
    float bi = s_im[t * PP + tid] + binI;
    float nr = fmaf(lr, xr, fmaf(-li, xi, br));
    float ni = fmaf(lr, xi, fmaf(li, xr, bi));
    xr = nr; xi = ni;
    s_re[t * PP + tid] = xr;
    s_im[t * PP + tid] = xi;
  }

  // ---- 4) cross-chunk carry chain (decoupled lookback) ----
  float cinr = 0.f, cini = 0.f;
  const size_t cb = ((size_t)b * NCHUNK + c) * PDIM;
  if (c > 0) {
    if (tid == 0) {
      while (__hip_atomic_load(&flags[b * NCHUNK + c - 1], __ATOMIC_ACQUIRE,
                               __HIP_MEMORY_SCOPE_AGENT) == 0)
        __builtin_amdgcn_s_sleep(1);
    }
    __syncthreads();
    cinr = carryRe[cb - PDIM + tid];
    cini = carryIm[cb - PDIM + tid];
  }
  {
    const float ltr = lamTRe[tid], lti = lamTIm[tid];  // lam_bar^TCH
    carryRe[cb + tid] = fmaf(ltr, cinr, fmaf(-lti, cini, xr));
    carryIm[cb + tid] = fmaf(ltr, cini, fmaf(lti, cinr, xi));
  }
  __threadfence();
  __syncthreads();
  if (tid == 0)
    __hip_atomic_store(&flags[b * NCHUNK + c], 1, __ATOMIC_RELEASE,
                       __HIP_MEMORY_SCOPE_AGENT);

  if (c > 0) {  // correction: x_t += lam_bar^{t+1} * carry_in
    float fr = lr, fi = li;
    for (int t = 0; t < TCH; ++t) {
      s_re[t * PP + tid] += fr * cinr - fi * cini;
      s_im[t * PP + tid] += fr * cini + fi * cinr;
      float nfr = fr * lr - fi * li;
      float nfi = fr * li + fi * lr;
      fr = nfr; fi = nfi;
    }
  }
  __syncthreads();

  // ---- 5) output projection + epilogue ----
  for (int job = wave; job < (TCH / 16) * OT; job += 8) {
    const int mt = job / OT;
    const int nt = job - mt * OT;
    const int o0 = nt * 16;
    const int arow = (mt * 16 + ln);
    v8f acc = zero8();
#pragma unroll 4
    for (int k = 0; k < PDIM / 4; ++k) {
      int q = k * 2 + hi;
      v2f ar = ld2s(&s_re[arow * PP + k * 4 + hi * 2]);
      v2f wr = ld2s(&CtPRe[q * (OPAD * 2) + (o0 + ln) * 2]);
      acc = wmma4(ar, wr, acc);
      v2f ai = ld2s(&s_im[arow * PP + k * 4 + hi * 2]);
      v2f wi = ld2s(&CtPIm[q * (OPAD * 2) + (o0 + ln) * 2]);
      acc = wmma4(ai, wi, acc);
    }
    const float bo = boutP[o0 + ln];
#pragma unroll
    for (int r = 0; r < 8; ++r)
      acc[r] = fmaxf(acc[r] + bo, 0.f) * 0.70710678118654752f;  // relu/sqrt2
    if (SKIP) {
#pragma unroll
      for (int k = 0; k < KH; ++k) {
        v2f a = ld2s(&s_u[arow * HPAD + k * 4 + hi * 2]);
        v2f w = ld2s(&skipWP[(k * 2 + hi) * (OPAD * 2) + (o0 + ln) * 2]);
        acc = wmma4(a, w, acc);
      }
      const float sb = skipBp[o0 + ln];
#pragma unroll
      for (int r = 0; r < 8; ++r) acc[r] += sb;
    }
    const int o = o0 + ln;
    if (o < O) {
      const int rb = mt * 16 + hi * 8;
#pragma unroll
      for (int r = 0; r < 8; ++r)
        xout[((size_t)b * LSEQ + t0 + rb + r) * O + o] = acc[r];
    }
  }
}

// ---------------- head: [8,16384,20] @ [20,1] + b ----------------
__global__ void head_kernel(const float* __restrict__ xin,
                            const float* __restrict__ headW,
                            const float* __restrict__ headb,
                            float* __restrict__ out) {
  int i = blockIdx.x * blockDim.x + threadIdx.x;
  if (i >= NB * LSEQ) return;
  const float* r = xin + (size_t)i * 20;
  float s = headb[0];
#pragma unroll
  for (int o = 0; o < 20; ++o) s = fmaf(r[o], headW[o], s);
  out[i] = s;
}

// ---------------- host side ----------------
extern "C" void kernel_launch(void* const* d_in, const int* in_sizes, int n_in,
                              void* d_out, int out_size, void* d_ws,
                              size_t ws_size, hipStream_t stream) {
  (void)in_sizes; (void)n_in; (void)out_size; (void)ws_size;

  // JAX pytree flatten order (dict keys sorted; ASCII: uppercase < lowercase):
  // midi; per layer: [skip_W, skip_b,] B_im, B_re, C_im, C_re, b_in, b_out,
  // lam_im, lam_re, log_step; head_W; head_b.
  struct LayerIn {
    const float *skip_W, *skip_b, *B_im, *B_re, *C_im, *C_re, *b_in, *b_out,
        *lam_im, *lam_re, *log_step;
  };
  const float* midi = (const float*)d_in[0];
  LayerIn L[4];
  int idx = 1;
  for (int l = 0; l < 4; ++l) {
    if (l > 0) {
      L[l].skip_W = (const float*)d_in[idx++];
      L[l].skip_b = (const float*)d_in[idx++];
    } else {
      L[l].skip_W = nullptr;
      L[l].skip_b = nullptr;
    }
    L[l].B_im = (const float*)d_in[idx++];
    L[l].B_re = (const float*)d_in[idx++];
    L[l].C_im = (const float*)d_in[idx++];
    L[l].C_re = (const float*)d_in[idx++];
    L[l].b_in = (const float*)d_in[idx++];
    L[l].b_out = (const float*)d_in[idx++];
    L[l].lam_im = (const float*)d_in[idx++];
    L[l].lam_re = (const float*)d_in[idx++];
    L[l].log_step = (const float*)d_in[idx++];
  }
  const float* head_W = (const float*)d_in[idx++];
  const float* head_b = (const float*)d_in[idx++];

  // ---- workspace partition (floats) ----
  float* ws = (float*)d_ws;
  size_t off = 0;
  auto alloc = [&](size_t n) {
    float* p = ws + off;
    off += (n + 63) & ~(size_t)63;
    return p;
  };
  float* xA = alloc((size_t)NB * LSEQ * 88);
  float* xB = alloc((size_t)NB * LSEQ * 88);
  int* flagsAll = (int*)alloc((size_t)4 * NB * NCHUNK);

  const int Hs[4] = {88, 88, 60, 40};
  const int Os[4] = {88, 60, 40, 20};
  const int OPs[4] = {96, 64, 48, 32};
  struct LayerWs {
    float *BtPRe, *BtPIm, *CtPRe, *CtPIm, *skipWP, *boutP, *skipBp;
    float *binBtRe, *binBtIm;
    float *lamRe, *lamIm, *lamTRe, *lamTIm, *carryRe, *carryIm;
    int* flags;
  } W[4];
  for (int l = 0; l < 4; ++l) {
    int H = Hs[l], OP = OPs[l];
    W[l].BtPRe = alloc((size_t)H * PDIM);
    W[l].BtPIm = alloc((size_t)H * PDIM);
    W[l].CtPRe = alloc((size_t)PDIM * OP);
    W[l].CtPIm = alloc((size_t)PDIM * OP);
    W[l].skipWP = alloc((size_t)H * OP);
    W[l].boutP = alloc(OP);
    W[l].skipBp = alloc(OP);
    W[l].binBtRe = alloc(PDIM);
    W[l].binBtIm = alloc(PDIM);
    W[l].lamRe = alloc(PDIM);
    W[l].lamIm = alloc(PDIM);
    W[l].lamTRe = alloc(PDIM);
    W[l].lamTIm = alloc(PDIM);
    W[l].carryRe = alloc((size_t)NB * NCHUNK * PDIM);
    W[l].carryIm = alloc((size_t)NB * NCHUNK * PDIM);
    W[l].flags = flagsAll + (size_t)l * NB * NCHUNK;
  }

  // flags must start at 0 each call (graph replays included)
  (void)hipMemsetAsync(flagsAll, 0, (size_t)4 * NB * NCHUNK * sizeof(int),
                       stream);

  for (int l = 0; l < 4; ++l) {
    int H = Hs[l], O = Os[l], OP = OPs[l];
    int jobs = PDIM + H * PDIM + PDIM * OP + H * OP + OP + PDIM;
    prep_kernel<<<(jobs + 255) / 256, 256, 0, stream>>>(
        L[l].B_im, L[l].B_re, L[l].C_im, L[l].C_re, L[l].b_out, L[l].lam_im,
        L[l].lam_re, L[l].log_step, L[l].skip_W, L[l].skip_b, L[l].b_in,
        W[l].BtPRe, W[l].BtPIm, W[l].CtPRe, W[l].CtPIm, W[l].skipWP,
        W[l].boutP, W[l].skipBp, W[l].binBtRe, W[l].binBtIm, W[l].lamRe,
        W[l].lamIm, W[l].lamTRe, W[l].lamTIm, H, O, OP, l > 0 ? 1 : 0);
  }

  dim3 grid(NB * NCHUNK), blk(256);
#define LAYER_ARGS(l, src, dst, O)                                           \
  src, dst, W[l].BtPRe, W[l].BtPIm, W[l].CtPRe, W[l].CtPIm, W[l].skipWP,     \
      W[l].boutP, W[l].skipBp, W[l].binBtRe, W[l].binBtIm, W[l].lamRe,       \
      W[l].lamIm, W[l].lamTRe, W[l].lamTIm, W[l].carryRe, W[l].carryIm,      \
      W[l].flags, O
  ssm_layer<88, 6, false, true><<<grid, blk, 0, stream>>>(
      LAYER_ARGS(0, midi, xA, 88));
  ssm_layer<88, 4, true, false><<<grid, blk, 0, stream>>>(
      LAYER_ARGS(1, xA, xB, 60));
  ssm_layer<60, 3, true, false><<<grid, blk, 0, stream>>>(
      LAYER_ARGS(2, xB, xA, 40));
  ssm_layer<40, 2, true, false><<<grid, blk, 0, stream>>>(
      LAYER_ARGS(3, xA, xB, 20));
#undef LAYER_ARGS

  head_kernel<<<(NB * LSEQ + 255) / 256, 256, 0, stream>>>(
      xB, head_W, head_b, (float*)d_out);
}